// GraphNet_24215025615450
// MI455X (gfx1250) — compile-verified
//
#include <hip/hip_runtime.h>
#include <math.h>

#define H 128
#define NEG_SLOPE 0.2f
#define NPART 8  // column partitions for the KNN GEMM (occupancy)

typedef float v2f __attribute__((ext_vector_type(2)));
typedef float v8f __attribute__((ext_vector_type(8)));

// ---------------------------------------------------------------------------
// WMMA f32 16x16x4 chained over K=128.
// A-fragment (16x4 f32): lane l holds row m=l&15, K pair base kb=(l>>4)*2.
// B-fragment (4x16 f32): lane l holds col n=l&15, same K pair base.
// C/D (16x16 f32, 8 VGPRs): vgpr i -> row i (lanes 0-15) / row i+8 (lanes 16-31).
// ---------------------------------------------------------------------------
__device__ __forceinline__ v8f wmma_acc_k128(const float* __restrict__ Arow,
                                             const float* __restrict__ Bcol,
                                             int bKs, int lane, v8f c) {
  const int kb = ((lane >> 4) & 1) * 2;
#pragma unroll
  for (int k = 0; k < H; k += 4) {
    v2f a, b;
    a.x = Arow[k + kb];
    a.y = Arow[k + kb + 1];
    b.x = Bcol[(size_t)(k + kb) * bKs];
    b.y = Bcol[(size_t)(k + kb + 1) * bKs];
    c = __builtin_amdgcn_wmma_f32_16x16x4_f32(false, a, false, b, (short)0, c,
                                              false, false);
  }
  return c;
}

// float atomic max via signed-max / unsigned-min trick (init to -inf bits)
__device__ __forceinline__ void atomicMaxF(float* addr, float v) {
  if (v >= 0.0f)
    atomicMax((int*)addr, __float_as_int(v));
  else
    atomicMin((unsigned int*)addr, __float_as_uint(v));
}

// ---------------------------------------------------------------------------
// C[N x 128] = A[N x 128] @ B[128 x 128], one wave per 16x16 tile.
// ---------------------------------------------------------------------------
__global__ void wmma_gemm_nk128(const float* __restrict__ A,
                                const float* __restrict__ B,
                                float* __restrict__ C, int N) {
  int wave = (blockIdx.x * blockDim.x + threadIdx.x) >> 5;
  int lane = threadIdx.x & 31;
  int tm = wave >> 3;  // row tile
  int tn = wave & 7;   // col tile (128/16 = 8)
  int row0 = tm * 16;
  if (row0 >= N) return;
  int m = row0 + (lane & 15);
  if (m >= N) m = N - 1;  // clamp (garbage rows never written)
  const float* Arow = A + (size_t)m * H;
  const float* Bcol = B + tn * 16 + (lane & 15);
  v8f z = {};
  v8f c = wmma_acc_k128(Arow, Bcol, H, lane, z);
  int col = tn * 16 + (lane & 15);
  int rbase = row0 + ((lane >> 4) & 1) * 8;
#pragma unroll
  for (int i = 0; i < 8; ++i) {
    int r = rbase + i;
    if (r < N) C[(size_t)r * H + col] = c[i];
  }
}

// h[n][t] = sum_f X[n][f] * W[f][t]   (tiny Fin = 3 or 6)
__global__ void xw_small(const float* __restrict__ X, const float* __restrict__ W,
                         float* __restrict__ h, int Fin) {
  int n = blockIdx.x, t = threadIdx.x;
  float acc = 0.0f;
  for (int f = 0; f < Fin; ++f) acc += X[(size_t)n * Fin + f] * W[f * H + t];
  h[(size_t)n * H + t] = acc;
}

// a_s[n] = dot(h[n], a_src); a_d[n] = dot(h[n], a_dst)
__global__ void node_scores(const float* __restrict__ h,
                            const float* __restrict__ asv,
                            const float* __restrict__ adv,
                            float* __restrict__ a_s, float* __restrict__ a_d) {
  __shared__ float ss[H];
  __shared__ float sd[H];
  int n = blockIdx.x, t = threadIdx.x;
  float hv = h[(size_t)n * H + t];
  ss[t] = hv * asv[t];
  sd[t] = hv * adv[t];
  __syncthreads();
  for (int s = H / 2; s > 0; s >>= 1) {
    if (t < s) { ss[t] += ss[t + s]; sd[t] += sd[t + s]; }
    __syncthreads();
  }
  if (t == 0) { a_s[n] = ss[0]; a_d[n] = sd[0]; }
}

__global__ void gat_init(float* __restrict__ out, float* __restrict__ e_max,
                         float* __restrict__ denom, int N) {
  int tid = blockIdx.x * blockDim.x + threadIdx.x;
  if (tid < N * H) out[tid] = 0.0f;
  if (tid < N) {
    ((int*)e_max)[tid] = 0xFF800000;  // -inf
    denom[tid] = 0.0f;
  }
}

__global__ void edge_logits_max(const int* __restrict__ edges, int E, int N,
                                const float* __restrict__ a_s,
                                const float* __restrict__ a_d,
                                float* __restrict__ e_val,
                                float* __restrict__ e_max) {
  int e = blockIdx.x * blockDim.x + threadIdx.x;
  if (e >= E + N) return;
  int src = (e < E) ? edges[e] : (e - E);
  int dst = (e < E) ? edges[E + e] : (e - E);
  float v = a_s[src] + a_d[dst];
  v = (v > 0.0f) ? v : v * NEG_SLOPE;
  e_val[e] = v;
  atomicMaxF(&e_max[dst], v);
}

__global__ void edge_exp_sum(const int* __restrict__ edges, int E, int N,
                             float* __restrict__ e_val,
                             const float* __restrict__ e_max,
                             float* __restrict__ denom) {
  int e = blockIdx.x * blockDim.x + threadIdx.x;
  if (e >= E + N) return;
  int dst = (e < E) ? edges[E + e] : (e - E);
  float ex = expf(e_val[e] - e_max[dst]);
  e_val[e] = ex;
  atomicAdd(&denom[dst], ex);
}

// one block (128 threads) per edge: out[dst] += alpha * h[src]
__global__ void edge_scatter(const int* __restrict__ edges, int E, int N,
                             const float* __restrict__ e_val,
                             const float* __restrict__ denom,
                             const float* __restrict__ h,
                             float* __restrict__ out) {
  int e = blockIdx.x;
  int t = threadIdx.x;
  int src = (e < E) ? edges[e] : (e - E);
  int dst = (e < E) ? edges[E + e] : (e - E);
  float alpha = e_val[e] / denom[dst];
  atomicAdd(&out[(size_t)dst * H + t], alpha * h[(size_t)src * H + t]);
}

__global__ void bias_relu(float* __restrict__ out, const float* __restrict__ b,
                          int total) {
  int tid = blockIdx.x * blockDim.x + threadIdx.x;
  if (tid >= total) return;
  float v = out[tid] + b[tid & (H - 1)];
  out[tid] = (v > 0.0f) ? v : 0.0f;
}

// rnorm[n] = 0.5 * |hr[n]|^2
__global__ void half_norm(const float* __restrict__ hr, float* __restrict__ rn) {
  __shared__ float ss[H];
  int n = blockIdx.x, t = threadIdx.x;
  float v = hr[(size_t)n * H + t];
  ss[t] = v * v;
  __syncthreads();
  for (int s = H / 2; s > 0; s >>= 1) {
    if (t < s) ss[t] += ss[t + s];
    __syncthreads();
  }
  if (t == 0) rn[n] = 0.5f * ss[0];
}

// ---------------------------------------------------------------------------
// KNN partial pass: grid = (colliderTiles, NPART). One wave per block.
// Each wave streams its partition's 16-wide resting-column tiles, computing
// score = hc.hr - 0.5|hr|^2 via WMMA, and keeps a per-row top-3 for its
// partition. Candidates go to workspace; knn_merge reduces across partitions.
// ---------------------------------------------------------------------------
__global__ void knn_topk_part(const float* __restrict__ hc,
                              const float* __restrict__ hr,
                              const float* __restrict__ rnorm,
                              float* __restrict__ cand_s,
                              int* __restrict__ cand_i, int Nc, int Nr) {
  __shared__ float sc[16][17];
  int lane = threadIdx.x;  // blockDim == 32
  int row0 = blockIdx.x * 16;
  int part = blockIdx.y;
  int m = row0 + (lane & 15);
  if (m >= Nc) m = Nc - 1;  // clamp loads; clamped rows never written out
  const float* Arow = hc + (size_t)m * H;

  int T = (Nr + 15) / 16;                 // total 16-col tiles
  int tpp = (T + NPART - 1) / NPART;      // tiles per partition
  int t0 = part * tpp;
  int t1 = (t0 + tpp < T) ? (t0 + tpp) : T;

  float b0 = -INFINITY, b1 = -INFINITY, b2 = -INFINITY;
  int i0 = 0x7fffffff, i1 = 0x7fffffff, i2 = 0x7fffffff;

  for (int tt = t0; tt < t1; ++tt) {
    int col0 = tt * 16;
    int col = col0 + (lane & 15);
    int colc = (col < Nr) ? col : (Nr - 1);
    const float* Bcol = hr + (size_t)colc * H;  // B[k][n] = hr[n][k]
    v8f z = {};
    v8f c = wmma_acc_k128(Arow, Bcol, 1, lane, z);
    float rn = rnorm[colc];
    int mrow = ((lane >> 4) & 1) * 8;
    bool cval = (col < Nr);
#pragma unroll
    for (int i = 0; i < 8; ++i)
      sc[mrow + i][lane & 15] = cval ? (c[i] - rn) : -INFINITY;
    __syncthreads();
    if (lane < 16) {
#pragma unroll
      for (int j = 0; j < 16; ++j) {
        float v = sc[lane][j];
        int idx = col0 + j;
        if (v > b0)      { b2 = b1; i2 = i1; b1 = b0; i1 = i0; b0 = v; i0 = idx; }
        else if (v > b1) { b2 = b1; i2 = i1; b1 = v; i1 = idx; }
        else if (v > b2) { b2 = v; i2 = idx; }
      }
    }
    __syncthreads();
  }
  int row = row0 + lane;
  if (lane < 16 && row < Nc) {
    size_t base = ((size_t)row * NPART + part) * 3;
    cand_s[base + 0] = b0; cand_i[base + 0] = i0;
    cand_s[base + 1] = b1; cand_i[base + 1] = i1;
    cand_s[base + 2] = b2; cand_i[base + 2] = i2;
  }
}

// merge NPART*3 candidates per collider row -> final top-3 indices.
// Comparator: higher score wins; on equal score lower index wins (top_k ties).
__global__ void knn_merge(const float* __restrict__ cand_s,
                          const int* __restrict__ cand_i,
                          int* __restrict__ kidx, int Nc) {
  int n = blockIdx.x * blockDim.x + threadIdx.x;
  if (n >= Nc) return;
  float b0 = -INFINITY, b1 = -INFINITY, b2 = -INFINITY;
  int i0 = 0x7fffffff, i1 = 0x7fffffff, i2 = 0x7fffffff;
  size_t base = (size_t)n * NPART * 3;
  for (int j = 0; j < NPART * 3; ++j) {
    float v = cand_s[base + j];
    int idx = cand_i[base + j];
    if (v > b0 || (v == b0 && idx < i0)) {
      b2 = b1; i2 = i1; b1 = b0; i1 = i0; b0 = v; i0 = idx;
    } else if (v > b1 || (v == b1 && idx < i1)) {
      b2 = b1; i2 = i1; b1 = v; i1 = idx;
    } else if (v > b2 || (v == b2 && idx < i2)) {
      b2 = v; i2 = idx;
    }
  }
  kidx[n * 3 + 0] = i0;
  kidx[n * 3 + 1] = i1;
  kidx[n * 3 + 2] = i2;
}

__global__ void pool_init(float* __restrict__ sums, float* __restrict__ cnt,
                          int Nr) {
  int tid = blockIdx.x * blockDim.x + threadIdx.x;
  if (tid < Nr * H) sums[tid] = 0.0f;
  if (tid < Nr) cnt[tid] = 0.0f;
}

// one block per knn edge (Nc*3): sums[r] += hc[c]; cnt[r] += 1
__global__ void knn_scatter(const int* __restrict__ knn_idx,
                            const float* __restrict__ hc,
                            float* __restrict__ sums, float* __restrict__ cnt) {
  int e = blockIdx.x;
  int t = threadIdx.x;
  int c = e / 3;
  int r = knn_idx[e];
  atomicAdd(&sums[(size_t)r * H + t], hc[(size_t)c * H + t]);
  if (t == 0) atomicAdd(&cnt[r], 1.0f);
}

// out[n] = concat(hr[n], sums[n]/max(cnt,1)) @ Wd + bd
__global__ void decoder(const float* __restrict__ hr,
                        const float* __restrict__ sums,
                        const float* __restrict__ cnt,
                        const float* __restrict__ Wd,
                        const float* __restrict__ bd,
                        float* __restrict__ out, int Nr) {
  int n = blockIdx.x * blockDim.x + threadIdx.x;
  if (n >= Nr) return;
  float inv = 1.0f / fmaxf(cnt[n], 1.0f);
  float a0 = bd[0], a1 = bd[1], a2 = bd[2];
  for (int k = 0; k < H; ++k) {
    float hv = hr[(size_t)n * H + k];
    a0 += hv * Wd[k * 3 + 0];
    a1 += hv * Wd[k * 3 + 1];
    a2 += hv * Wd[k * 3 + 2];
    float pv = sums[(size_t)n * H + k] * inv;
    a0 += pv * Wd[(H + k) * 3 + 0];
    a1 += pv * Wd[(H + k) * 3 + 1];
    a2 += pv * Wd[(H + k) * 3 + 2];
  }
  out[n * 3 + 0] = a0;
  out[n * 3 + 1] = a1;
  out[n * 3 + 2] = a2;
}

// ---------------------------------------------------------------------------
static void run_gat_layer(const float* X, int N, int Fin, const int* edges,
                          int E, const float* W, const float* asv,
                          const float* adv, const float* bias, float* h,
                          float* out, float* a_s, float* a_d, float* e_max,
                          float* denom, float* e_val, hipStream_t stream) {
  if (Fin == H) {
    int waves = ((N + 15) / 16) * 8;
    int blocks = (waves + 7) / 8;  // 8 waves (256 thr) per block
    wmma_gemm_nk128<<<blocks, 256, 0, stream>>>(X, W, h, N);
  } else {
    xw_small<<<N, H, 0, stream>>>(X, W, h, Fin);
  }
  node_scores<<<N, H, 0, stream>>>(h, asv, adv, a_s, a_d);
  gat_init<<<(N * H + 255) / 256, 256, 0, stream>>>(out, e_max, denom, N);
  int Etot = E + N;
  edge_logits_max<<<(Etot + 255) / 256, 256, 0, stream>>>(edges, E, N, a_s,
                                                          a_d, e_val, e_max);
  edge_exp_sum<<<(Etot + 255) / 256, 256, 0, stream>>>(edges, E, N, e_val,
                                                       e_max, denom);
  edge_scatter<<<Etot, H, 0, stream>>>(edges, E, N, e_val, denom, h, out);
  bias_relu<<<(N * H + 255) / 256, 256, 0, stream>>>(out, bias, N * H);
}

extern "C" void kernel_launch(void* const* d_in, const int* in_sizes, int n_in,
                              void* d_out, int out_size, void* d_ws,
                              size_t ws_size, hipStream_t stream) {
  const float* x_r  = (const float*)d_in[0];
  const int*   e_r  = (const int*)d_in[1];
  const float* x_c  = (const float*)d_in[2];
  const int*   e_c  = (const int*)d_in[3];
  const float* W1r  = (const float*)d_in[4];
  const float* as1r = (const float*)d_in[5];
  const float* ad1r = (const float*)d_in[6];
  const float* b1r  = (const float*)d_in[7];
  const float* W2r  = (const float*)d_in[8];
  const float* as2r = (const float*)d_in[9];
  const float* ad2r = (const float*)d_in[10];
  const float* b2r  = (const float*)d_in[11];
  const float* W1c  = (const float*)d_in[12];
  const float* as1c = (const float*)d_in[13];
  const float* ad1c = (const float*)d_in[14];
  const float* b1c  = (const float*)d_in[15];
  const float* W2c  = (const float*)d_in[16];
  const float* as2c = (const float*)d_in[17];
  const float* ad2c = (const float*)d_in[18];
  const float* b2c  = (const float*)d_in[19];
  const float* Wd   = (const float*)d_in[20];
  const float* bd   = (const float*)d_in[21];
  float* out = (float*)d_out;

  const int Nr = in_sizes[0] / 3;
  const int Er = in_sizes[1] / 2;
  const int Nc = in_sizes[2] / 6;
  const int Ec = in_sizes[3] / 2;
  const int Nmax = (Nr > Nc) ? Nr : Nc;
  const int Emax = (Er > Ec) ? Er : Ec;

  // carve workspace
  char* p = (char*)d_ws;
  auto carve = [&](size_t bytes) -> char* {
    char* r = p;
    p += (bytes + 255) & ~(size_t)255;
    return r;
  };
  float* h_tmp = (float*)carve((size_t)Nmax * H * 4);  // pre-agg h per layer
  float* bufA  = (float*)carve((size_t)Nr * H * 4);    // resting layer1 out
  float* bufB  = (float*)carve((size_t)Nr * H * 4);    // resting layer2 out
  float* cbufA = (float*)carve((size_t)Nc * H * 4);    // collider layer1 out
  float* cbufB = (float*)carve((size_t)Nc * H * 4);    // collider layer2 out
  float* a_s   = (float*)carve((size_t)Nmax * 4);
  float* a_d   = (float*)carve((size_t)Nmax * 4);
  float* e_max = (float*)carve((size_t)Nmax * 4);
  float* denom = (float*)carve((size_t)Nmax * 4);
  float* e_val = (float*)carve((size_t)(Emax + Nmax) * 4);
  float* rnorm = (float*)carve((size_t)Nr * 4);
  float* sums  = (float*)carve((size_t)Nr * H * 4);
  float* cnt   = (float*)carve((size_t)Nr * 4);
  int*   kidx  = (int*)carve((size_t)Nc * 3 * 4);
  float* cand_s = (float*)carve((size_t)Nc * NPART * 3 * 4);
  int*   cand_i = (int*)carve((size_t)Nc * NPART * 3 * 4);
  (void)ws_size; (void)n_in; (void)out_size;

  // resting branch
  run_gat_layer(x_r, Nr, 3, e_r, Er, W1r, as1r, ad1r, b1r, h_tmp, bufA, a_s,
                a_d, e_max, denom, e_val, stream);
  run_gat_layer(bufA, Nr, H, e_r, Er, W2r, as2r, ad2r, b2r, h_tmp, bufB, a_s,
                a_d, e_max, denom, e_val, stream);
  // collider branch
  run_gat_layer(x_c, Nc, 6, e_c, Ec, W1c, as1c, ad1c, b1c, h_tmp, cbufA, a_s,
                a_d, e_max, denom, e_val, stream);
  run_gat_layer(cbufA, Nc, H, e_c, Ec, W2c, as2c, ad2c, b2c, h_tmp, cbufB, a_s,
                a_d, e_max, denom, e_val, stream);

  // knn pool: hr = bufB, hc = cbufB
  half_norm<<<Nr, H, 0, stream>>>(bufB, rnorm);
  dim3 kgrid((Nc + 15) / 16, NPART);
  knn_topk_part<<<kgrid, 32, 0, stream>>>(cbufB, bufB, rnorm, cand_s, cand_i,
                                          Nc, Nr);
  knn_merge<<<(Nc + 127) / 128, 128, 0, stream>>>(cand_s, cand_i, kidx, Nc);
  pool_init<<<(Nr * H + 255) / 256, 256, 0, stream>>>(sums, cnt, Nr);
  knn_scatter<<<Nc * 3, H, 0, stream>>>(kidx, cbufB, sums, cnt);

  // decoder
  decoder<<<(Nr + 127) / 128, 128, 0, stream>>>(bufB, sums, cnt, Wd, bd, out,
                                                Nr);
}